// GRUGNNDecoder_48859547959740
// MI455X (gfx1250) — compile-verified
//
#include <hip/hip_runtime.h>
#include <math.h>

// ---------------------------------------------------------------------------
// GRU-GNN decoder step, MI455X (gfx1250, wave32, WMMA f32 16x16x4)
// B=50000 nodes (3125 tiles of 16), H=64, E=800000 edges, L=10, M=6
// ---------------------------------------------------------------------------

#define NB      50000
#define NE      800000
#define HID     64
#define LSTEPS  10
#define NMIX    6
#define LD      66          // padded LDS row stride (floats) to avoid bank conflicts
#define NTILES  3125        // NB / 16

typedef float v2f __attribute__((ext_vector_type(2)));
typedef float v8f __attribute__((ext_vector_type(8)));

// V_WMMA_F32_16X16X4_F32 : D(16x16 f32) = A(16x4 f32) * B(4x16 f32) + C
__device__ __forceinline__ v8f wmma4(v2f a, v2f b, v8f c) {
    return __builtin_amdgcn_wmma_f32_16x16x4_f32(false, a, false, b, (short)0, c,
                                                 false, false);
}

// A-fragment: 16x4 f32.  lanes 0-15: M=lane, K=4c+{0,1}; lanes 16-31: M=lane-16, K=4c+{2,3}
__device__ __forceinline__ v2f a_frag(const float* Xl, int c, int lane) {
    int row = lane & 15;
    int k   = (c << 2) + ((lane >> 4) << 1);
    const float* p = Xl + row * LD + k;
    v2f a; a.x = p[0]; a.y = p[1];
    return a;
}

// B-fragment: 4x16 f32 from row-major W[K x ncols] (ld = ldw), tile t over N.
__device__ __forceinline__ v2f b_frag(const float* W, int ldw, int ncols,
                                      int c, int t, int lane) {
    int n = (lane & 15) + (t << 4);
    int k = (c << 2) + ((lane >> 4) << 1);
    v2f b;
    if (n < ncols) { b.x = W[k * ldw + n]; b.y = W[(k + 1) * ldw + n]; }
    else           { b.x = 0.f;            b.y = 0.f; }
    return b;
}

// Generic 16-row GEMM: Y = act( [X0|X1] @ [W0;W1] + bias0 + bias1 ) (+ addl residual)
// act: 0=none, 1=leaky_relu(0.2), 2=tanh, 3=silu.  Writes to LDS (Yl) and/or global (G).
__device__ __forceinline__ void gemm16(
    const float* Xl0, int kch0, const float* W0,
    const float* Xl1, int kch1, const float* W1,
    int ldw, int ncols, int ntiles,
    const float* bias0, const float* bias1,
    int act,
    float* Yl, float* G, int ldg,
    const float* addl, int lane)
{
    int n15 = lane & 15;
    int hi  = (lane >> 4) << 3;
    for (int t = 0; t < ntiles; ++t) {
        int n = n15 + (t << 4);
        float bv = 0.f;
        if (n < ncols) {
            if (bias0) bv += bias0[n];
            if (bias1) bv += bias1[n];
        }
        v8f acc;
        #pragma unroll
        for (int r = 0; r < 8; ++r) acc[r] = bv;
        for (int c = 0; c < kch0; ++c)
            acc = wmma4(a_frag(Xl0, c, lane), b_frag(W0, ldw, ncols, c, t, lane), acc);
        if (Xl1)
            for (int c = 0; c < kch1; ++c)
                acc = wmma4(a_frag(Xl1, c, lane), b_frag(W1, ldw, ncols, c, t, lane), acc);
        #pragma unroll
        for (int r = 0; r < 8; ++r) {
            float v = acc[r];
            if      (act == 1) v = v > 0.f ? v : 0.2f * v;
            else if (act == 2) v = tanhf(v);
            else if (act == 3) v = v / (1.f + __expf(-v));
            int m = r + hi;
            if (addl) v += addl[m * LD + n];
            if (Yl) Yl[m * LD + n] = v;
            if (G && n < ncols) G[(size_t)m * ldg + n] = v;
        }
    }
}

__device__ __forceinline__ float softplus_clip(float x) {
    float y = (x > 20.f) ? x : log1pf(__expf(x));
    return fminf(fmaxf(y, 0.f), 100.f);
}

__device__ __forceinline__ void atomAddF(float* p, float v) {
    unsafeAtomicAdd(p, v);   // global_atomic_add_f32
}

// ---------------------------------------------------------------------------
// Kernel 1: fold linear heads.
//   Whead[64x32]: cols 0-5  = W_gen[:,0:64]   @ W_s1
//                 cols 6-11 = W_gen[:,64:128] @ W_s2
//                 cols12-17 = W_gen[:,128:192]@ W_rho
//                 cols18-29 = mean_k W_ctrl reshaped (H,M,NU,128)
//   bhead matching biases.
// ---------------------------------------------------------------------------
__global__ void k_prep(const float* W_gen, const float* b_gen,
                       const float* W_ctrl, const float* b_ctrl,
                       const float* W_s1, const float* b_s1,
                       const float* W_s2, const float* b_s2,
                       const float* W_rho, const float* b_rho,
                       float* Whead, float* bhead)
{
    int t = blockIdx.x * blockDim.x + threadIdx.x;
    int nthr = gridDim.x * blockDim.x;
    for (int idx = t; idx < 64 * 32; idx += nthr) {
        int i = idx >> 5, j = idx & 31;
        float s = 0.f;
        if (j < 6)       { for (int k = 0; k < 64; ++k)  s += W_gen[i * 192 + k]       * W_s1[k * 6 + j]; }
        else if (j < 12) { int jj = j - 6;
                           for (int k = 0; k < 64; ++k)  s += W_gen[i * 192 + 64 + k]  * W_s2[k * 6 + jj]; }
        else if (j < 18) { int jj = j - 12;
                           for (int k = 0; k < 64; ++k)  s += W_gen[i * 192 + 128 + k] * W_rho[k * 6 + jj]; }
        else if (j < 30) { int jj = j - 18; float a = 0.f;
                           for (int k = 0; k < 128; ++k) a += W_ctrl[(size_t)i * 1536 + jj * 128 + k];
                           s = a * (1.f / 128.f); }
        Whead[i * 32 + j] = s;
    }
    if (t < 32) {
        int j = t; float s = 0.f;
        if (j < 6)       { for (int k = 0; k < 64; ++k)  s += b_gen[k]       * W_s1[k * 6 + j];  s += b_s1[j]; }
        else if (j < 12) { int jj = j - 6;
                           for (int k = 0; k < 64; ++k)  s += b_gen[64 + k]  * W_s2[k * 6 + jj]; s += b_s2[jj]; }
        else if (j < 18) { int jj = j - 12;
                           for (int k = 0; k < 64; ++k)  s += b_gen[128 + k] * W_rho[k * 6 + jj]; s += b_rho[jj]; }
        else if (j < 30) { int jj = j - 18; float a = 0.f;
                           for (int k = 0; k < 128; ++k) a += b_ctrl[jj * 128 + k];
                           s = a * (1.f / 128.f); }
        bhead[j] = s;
    }
}

__global__ void k_zero(float* p, int n)
{
    int t = blockIdx.x * blockDim.x + threadIdx.x;
    if (t < n) p[t] = 0.f;
}

// ---------------------------------------------------------------------------
// Kernel 2: per-node "pre" stage.  One wave (32 threads) per 16-node tile.
//   emb, attention softmax + encoder einsum, comb, GraphConv pre-terms,
//   g-MLP (tanh^3) and hi-MLP (silu^3).
// ---------------------------------------------------------------------------
__global__ __launch_bounds__(32)
void k_node_pre(const float* __restrict__ hidden, const float* __restrict__ enc,
                const float* __restrict__ past,
                const float* W_emb, const float* b_emb,
                const float* W_attn, const float* b_attn,
                const float* W_comb, const float* b_comb,
                const float* W_root, const float* W_nbr,
                const float* b_gc, const float* bias_g,
                const float* Whi1, const float* bhi1,
                const float* Whi2, const float* bhi2,
                const float* Whi3, const float* bhi3,
                const float* Wg1, const float* bg1,
                const float* Wg2, const float* bg2,
                const float* Wg3, const float* bg3,
                float* __restrict__ nbr, float* __restrict__ prewrh,
                float* __restrict__ wrrb, float* __restrict__ hahab)
{
    __shared__ float T0[16 * LD], T1[16 * LD], T2[16 * LD], T3[16 * LD];
    int lane = threadIdx.x & 31;
    int r0   = blockIdx.x << 4;

    // past_state tile (16 x 24) -> T0
    for (int i = lane; i < 16 * 24; i += 32) {
        int r = i / 24, k = i - 24 * r;
        T0[r * LD + k] = past[(size_t)(r0 + r) * 24 + k];
    }
    __syncthreads();

    // emb = past @ W_emb + b_emb  -> T1      (K=24 -> 6 chunks)
    gemm16(T0, 6, W_emb, nullptr, 0, nullptr, 64, 64, 4,
           b_emb, nullptr, 0, T1, nullptr, 0, nullptr, lane);
    __syncthreads();

    // hidden tile -> T0
    for (int i = lane; i < 16 * 64; i += 32) {
        int r = i >> 6, k = i & 63;
        T0[r * LD + k] = hidden[(size_t)(r0 + r) * 64 + k];
    }
    __syncthreads();

    // logits = [emb|hidden] @ W_attn + b_attn -> T2   (N=10, one tile)
    gemm16(T1, 16, W_attn, T0, 16, W_attn + 64 * LSTEPS, LSTEPS, LSTEPS, 1,
           b_attn, nullptr, 0, T2, nullptr, 0, nullptr, lane);
    __syncthreads();

    // softmax over L=10 per row (lanes 0..15 own rows)
    if (lane < 16) {
        float mx = -1e30f;
        for (int l = 0; l < LSTEPS; ++l) mx = fmaxf(mx, T2[lane * LD + l]);
        float e[LSTEPS], s = 0.f;
        for (int l = 0; l < LSTEPS; ++l) { e[l] = __expf(T2[lane * LD + l] - mx); s += e[l]; }
        float inv = 1.f / s;
        for (int l = 0; l < LSTEPS; ++l) T2[lane * LD + l] = e[l] * inv;
    }
    __syncthreads();

    // attn_applied = sum_l w[l] * enc[:,l,:]  -> T3   (streams the 128MB tensor)
    for (int j = 0; j < 32; ++j) {
        int r = j >> 1, h = ((j & 1) << 5) + lane;
        const float* ep = enc + (size_t)(r0 + r) * (LSTEPS * 64) + h;
        float s = 0.f;
        #pragma unroll
        for (int l = 0; l < LSTEPS; ++l) s += T2[r * LD + l] * ep[l * 64];
        T3[r * LD + h] = s;
    }
    __syncthreads();

    // out = leaky_relu([emb|attn] @ W_comb + b_comb) -> T2
    gemm16(T1, 16, W_comb, T3, 16, W_comb + 64 * 64, 64, 64, 4,
           b_comb, nullptr, 1, T2, nullptr, 0, nullptr, lane);
    __syncthreads();

    // nbr = out @ W_nbr -> global
    gemm16(T2, 16, W_nbr, nullptr, 0, nullptr, 64, 64, 4,
           nullptr, nullptr, 0, nullptr, nbr + (size_t)r0 * 64, 64, nullptr, lane);
    // prewrh = bias_g + b_gc + out @ W_root -> global
    gemm16(T2, 16, W_root, nullptr, 0, nullptr, 64, 64, 4,
           bias_g, b_gc, 0, nullptr, prewrh + (size_t)r0 * 64, 64, nullptr, lane);

    // wrr = tanh-MLP3(out) -> global
    gemm16(T2, 16, Wg1, nullptr, 0, nullptr, 64, 64, 4,
           bg1, nullptr, 2, T1, nullptr, 0, nullptr, lane);
    __syncthreads();
    gemm16(T1, 16, Wg2, nullptr, 0, nullptr, 64, 64, 4,
           bg2, nullptr, 2, T3, nullptr, 0, nullptr, lane);
    __syncthreads();
    gemm16(T3, 16, Wg3, nullptr, 0, nullptr, 64, 64, 4,
           bg3, nullptr, 2, nullptr, wrrb + (size_t)r0 * 64, 64, nullptr, lane);

    // haha = silu-MLP3(hidden) -> global   (hidden still lives in T0)
    gemm16(T0, 16, Whi1, nullptr, 0, nullptr, 64, 64, 4,
           bhi1, nullptr, 3, T1, nullptr, 0, nullptr, lane);
    __syncthreads();
    gemm16(T1, 16, Whi2, nullptr, 0, nullptr, 64, 64, 4,
           bhi2, nullptr, 3, T3, nullptr, 0, nullptr, lane);
    __syncthreads();
    gemm16(T3, 16, Whi3, nullptr, 0, nullptr, 64, 64, 4,
           bhi3, nullptr, 3, nullptr, hahab + (size_t)r0 * 64, 64, nullptr, lane);
}

// ---------------------------------------------------------------------------
// Kernel 3: edge scatter  agg[dst] += w * nbr[src]   (16 threads / edge, float4)
// ---------------------------------------------------------------------------
__global__ void k_edges(const int* __restrict__ eidx, const float* __restrict__ ew,
                        const float* __restrict__ nbr, float* __restrict__ agg)
{
    int t = blockIdx.x * blockDim.x + threadIdx.x;
    int e = t >> 4;
    if (e >= NE) return;
    int part = t & 15;
    int src = eidx[e];
    int dst = eidx[NE + e];
    float w = ew[e];
    const float4 v = ((const float4*)(nbr + (size_t)src * 64))[part];
    float* d = agg + (size_t)dst * 64 + part * 4;
    atomAddF(d + 0, w * v.x);
    atomAddF(d + 1, w * v.y);
    atomAddF(d + 2, w * v.z);
    atomAddF(d + 3, w * v.w);
}

// ---------------------------------------------------------------------------
// Kernel 4: per-node "post": GRU combine + LN + residual update + heads.
// ---------------------------------------------------------------------------
__global__ __launch_bounds__(32)
void k_node_post(const float* __restrict__ hidden, const float* __restrict__ past,
                 const float* __restrict__ prewrh, const float* __restrict__ agg,
                 const float* __restrict__ wrrb, const float* __restrict__ hahab,
                 const float* W_lin1, const float* b_lin1,
                 const float* W_lin2, const float* b_lin2,
                 const float* ln_g, const float* ln_b,
                 const float* Whead, const float* bhead,
                 float* __restrict__ o_ns, float* __restrict__ o_u,
                 float* __restrict__ o_q, float* __restrict__ o_h)
{
    __shared__ float T0[16 * LD], T1[16 * LD], T2[16 * LD], T3[16 * LD];
    __shared__ float mrow[16], rrow[16];
    int lane = threadIdx.x & 31;
    int r0   = blockIdx.x << 4;

    // v = (wrr + tanh(prewrh + agg)) * haha -> T0
    for (int i = lane; i < 16 * 64; i += 32) {
        int r = i >> 6, k = i & 63;
        size_t g = (size_t)(r0 + r) * 64 + k;
        float wrh = tanhf(prewrh[g] + agg[g]);
        T0[r * LD + k] = (wrrb[g] + wrh) * hahab[g];
    }
    __syncthreads();

    // LayerNorm statistics per row
    if (lane < 16) {
        float s = 0.f, s2 = 0.f;
        for (int k = 0; k < 64; ++k) { float v = T0[lane * LD + k]; s += v; s2 += v * v; }
        float m = s * (1.f / 64.f);
        float var = s2 * (1.f / 64.f) - m * m;
        mrow[lane] = m;
        rrow[lane] = rsqrtf(var + 1e-5f);
    }
    __syncthreads();

    // t = tanh(LN(v)) -> T1
    for (int i = lane; i < 16 * 64; i += 32) {
        int r = i >> 6, k = i & 63;
        T1[r * LD + k] = tanhf((T0[r * LD + k] - mrow[r]) * rrow[r] * ln_g[k] + ln_b[k]);
    }
    __syncthreads();

    // tl = t @ W_lin1 + b_lin1 -> T2
    gemm16(T1, 16, W_lin1, nullptr, 0, nullptr, 64, 64, 4,
           b_lin1, nullptr, 0, T2, nullptr, 0, nullptr, lane);

    // hidden tile -> T3
    for (int i = lane; i < 16 * 64; i += 32) {
        int r = i >> 6, k = i & 63;
        T3[r * LD + k] = hidden[(size_t)(r0 + r) * 64 + k];
    }
    __syncthreads();

    // h_new = hidden + hidden@W_lin2[0:64] + tl@W_lin2[64:128] + b_lin2 -> T0 & global
    gemm16(T3, 16, W_lin2, T2, 16, W_lin2 + 64 * 64, 64, 64, 4,
           b_lin2, nullptr, 0, T0, o_h + (size_t)r0 * 64, 64, T3, lane);
    __syncthreads();

    // heads = h_new @ Whead + bhead -> T1   (64x32 folded head matrix, 2 N tiles)
    gemm16(T0, 16, Whead, nullptr, 0, nullptr, 32, 32, 2,
           bhead, nullptr, 0, T1, nullptr, 0, nullptr, lane);
    __syncthreads();

    // finalize mixtures + motion model (lanes 0..15 own rows)
    if (lane < 16) {
        int row = lane;
        size_t gb = (size_t)(r0 + row);
        for (int m = 0; m < NMIX; ++m) {
            float s1  = softplus_clip(T1[row * LD + m]);
            float s2  = softplus_clip(T1[row * LD + 6 + m]);
            float rh  = T1[row * LD + 12 + m];
            rh = rh / (1.f + fabsf(rh));                    // soft_sign
            float cov = s1 * s2 * rh;
            float u0 = T1[row * LD + 18 + 2 * m];
            float u1 = T1[row * LD + 19 + 2 * m];
            const float* ps = past + gb * 24 + m * 4;
            float p0 = ps[0], p1 = ps[1], v0 = ps[2], v1 = ps[3];
            float* ns = o_ns + gb * 24 + m * 4;
            ns[0] = p0 + 0.1f * v0;  ns[1] = p1 + 0.1f * v1;
            ns[2] = v0 + 0.1f * u0;  ns[3] = v1 + 0.1f * u1;
            float* up = o_u + gb * 12 + m * 2;
            up[0] = u0; up[1] = u1;
            float* qp = o_q + gb * 24 + m * 4;
            qp[0] = s1 * s1; qp[1] = cov; qp[2] = cov; qp[3] = s2 * s2;
        }
    }
}

// ---------------------------------------------------------------------------
extern "C" void kernel_launch(void* const* d_in, const int* in_sizes, int n_in,
                              void* d_out, int out_size, void* d_ws, size_t ws_size,
                              hipStream_t stream)
{
    (void)in_sizes; (void)n_in; (void)out_size; (void)ws_size;

    const float* hidden  = (const float*)d_in[1];
    const float* enc     = (const float*)d_in[2];
    const int*   eidx    = (const int*)  d_in[3];
    const float* ew      = (const float*)d_in[4];
    const float* past    = (const float*)d_in[5];
    const float* W_emb   = (const float*)d_in[7];
    const float* b_emb   = (const float*)d_in[8];
    const float* W_attn  = (const float*)d_in[9];
    const float* b_attn  = (const float*)d_in[10];
    const float* W_comb  = (const float*)d_in[11];
    const float* b_comb  = (const float*)d_in[12];
    const float* W_root  = (const float*)d_in[13];
    const float* W_nbr   = (const float*)d_in[14];
    const float* b_gc    = (const float*)d_in[15];
    const float* bias_g  = (const float*)d_in[16];
    const float* Whi1 = (const float*)d_in[17]; const float* bhi1 = (const float*)d_in[18];
    const float* Whi2 = (const float*)d_in[19]; const float* bhi2 = (const float*)d_in[20];
    const float* Whi3 = (const float*)d_in[21]; const float* bhi3 = (const float*)d_in[22];
    const float* Wg1  = (const float*)d_in[23]; const float* bg1  = (const float*)d_in[24];
    const float* Wg2  = (const float*)d_in[25]; const float* bg2  = (const float*)d_in[26];
    const float* Wg3  = (const float*)d_in[27]; const float* bg3  = (const float*)d_in[28];
    const float* W_lin1 = (const float*)d_in[29]; const float* b_lin1 = (const float*)d_in[30];
    const float* W_lin2 = (const float*)d_in[31]; const float* b_lin2 = (const float*)d_in[32];
    const float* ln_g = (const float*)d_in[33]; const float* ln_b = (const float*)d_in[34];
    const float* W_gen  = (const float*)d_in[35]; const float* b_gen  = (const float*)d_in[36];
    const float* W_ctrl = (const float*)d_in[37]; const float* b_ctrl = (const float*)d_in[38];
    const float* W_s1 = (const float*)d_in[39]; const float* b_s1 = (const float*)d_in[40];
    const float* W_s2 = (const float*)d_in[41]; const float* b_s2 = (const float*)d_in[42];
    const float* W_rho = (const float*)d_in[43]; const float* b_rho = (const float*)d_in[44];

    // workspace layout (floats): five B*64 buffers + folded head weights
    float* ws      = (float*)d_ws;
    const size_t BH = (size_t)NB * 64;            // 3.2M floats each
    float* nbr     = ws;
    float* prewrh  = ws + 1 * BH;
    float* wrrb    = ws + 2 * BH;
    float* hahab   = ws + 3 * BH;
    float* agg     = ws + 4 * BH;
    float* Whead   = ws + 5 * BH;
    float* bhead   = Whead + 64 * 32;

    // outputs concatenated flat: next_state[B,6,4], u[B,6,2], q[B,6,2,2], h_new[B,64]
    float* out  = (float*)d_out;
    float* o_ns = out;
    float* o_u  = out + (size_t)NB * 24;
    float* o_q  = out + (size_t)NB * 36;
    float* o_h  = out + (size_t)NB * 60;

    k_prep<<<8, 256, 0, stream>>>(W_gen, b_gen, W_ctrl, b_ctrl,
                                  W_s1, b_s1, W_s2, b_s2, W_rho, b_rho,
                                  Whead, bhead);

    k_zero<<<(int)((BH + 255) / 256), 256, 0, stream>>>(agg, (int)BH);

    k_node_pre<<<NTILES, 32, 0, stream>>>(hidden, enc, past,
                                          W_emb, b_emb, W_attn, b_attn,
                                          W_comb, b_comb, W_root, W_nbr,
                                          b_gc, bias_g,
                                          Whi1, bhi1, Whi2, bhi2, Whi3, bhi3,
                                          Wg1, bg1, Wg2, bg2, Wg3, bg3,
                                          nbr, prewrh, wrrb, hahab);

    k_edges<<<(NE * 16 + 255) / 256, 256, 0, stream>>>(eidx, ew, nbr, agg);

    k_node_post<<<NTILES, 32, 0, stream>>>(hidden, past, prewrh, agg, wrrb, hahab,
                                           W_lin1, b_lin1, W_lin2, b_lin2,
                                           ln_g, ln_b, Whead, bhead,
                                           o_ns, o_u, o_q, o_h);
}